// LJ_39539468927522
// MI455X (gfx1250) — compile-verified
//
#include <hip/hip_runtime.h>
#include <stdint.h>

// ---------------------------------------------------------------------------
// LJ pair energy + segment-sum scatter, MI455X (gfx1250).
// Memory-bound streaming kernel: async global->LDS double-buffered pipeline
// (ASYNCcnt) + fire-and-forget global_atomic_add_f32 scatter.
// Roofline: 6.4M pairs * 12B = 77MB reads @ 23.3 TB/s ~= 3.3us floor; output
// (400KB) accumulates in L2 (192MB). Purely bandwidth-bound -> async-LDS
// prefetch pipeline, wide b128 staging, no-return f32 atomics.
// ---------------------------------------------------------------------------

#define WAVES_PER_BLOCK 8
#define LANES 32
#define PAIRS_PER_LANE 4
#define TILE_PAIRS (LANES * PAIRS_PER_LANE) // 128 pairs per wave-iteration

// e0 = 4*eps*((sigma/rc)^12 - (sigma/rc)^6), sigma=eps=1, rc=3
__host__ __device__ constexpr float lj_e0() {
    double sr  = 1.0 / 3.0;
    double sr2 = sr * sr;
    double sr6 = sr2 * sr2 * sr2;
    return (float)(4.0 * (sr6 * sr6 - sr6));
}

// Async copy of 16B/lane from global to LDS (CDNA5 async-LDS path, ASYNCcnt).
__device__ __forceinline__ void async_copy_b128(unsigned lds_addr, const void* gaddr) {
    asm volatile("global_load_async_to_lds_b128 %0, %1, off"
                 :
                 : "v"(lds_addr), "v"((unsigned long long)gaddr)
                 : "memory");
}

#define ASYNC_WAIT(N) asm volatile("s_wait_asynccnt %0" :: "n"(N) : "memory")

__global__ __launch_bounds__(256) void lj_scatter_kernel(
    const float* __restrict__ dist,
    const int*   __restrict__ ind2,   // [npairs, 2] row-major, col 0 used
    float*       __restrict__ out,
    int npairs, int ntiles)
{
    __shared__ float4 sd[WAVES_PER_BLOCK][2][LANES];      //  8 KB: 128 dists/tile
    __shared__ int4   si[WAVES_PER_BLOCK][2][2 * LANES];  // 16 KB: 256 idx ints/tile

    const int lane       = threadIdx.x & 31;
    const int w          = threadIdx.x >> 5;
    const int wavesTotal = gridDim.x * WAVES_PER_BLOCK;
    const int waveId     = blockIdx.x * WAVES_PER_BLOCK + w;

    const float e0 = lj_e0();

    auto body = [&](float d, int i0) {
        float x   = 1.0f / d;            // sigma == 1
        float x2  = x * x;
        float c6  = x2 * x2 * x2;
        float en  = 4.0f * (c6 * c6 - c6) - e0;   // eps == 1
        // no-return f32 atomic add -> global_atomic_add_f32 (STOREcnt tracked)
        __hip_atomic_fetch_add(&out[i0], 0.5f * en,
                               __ATOMIC_RELAXED, __HIP_MEMORY_SCOPE_AGENT);
    };

    auto stage = [&](int buf, int t) {
        const int base = t * TILE_PAIRS;
        // dist tile: 128 floats, lane L copies dist[base + 4L .. 4L+3]
        async_copy_b128((unsigned)(size_t)&sd[w][buf][lane],
                        dist + base + 4 * lane);
        // index slab: ints [2*base, 2*base+256), two b128 sweeps
        async_copy_b128((unsigned)(size_t)&si[w][buf][lane],
                        ind2 + 2 * base + 4 * lane);
        async_copy_b128((unsigned)(size_t)&si[w][buf][LANES + lane],
                        ind2 + 2 * base + 128 + 4 * lane);
    };

    int t   = waveId;
    int buf = 0;
    if (t < ntiles) stage(0, t);                 // prime the pipeline

    for (; t < ntiles; t += wavesTotal) {
        const int nt = t + wavesTotal;
        if (nt < ntiles) {
            stage(buf ^ 1, nt);                  // prefetch next tile (async)
            ASYNC_WAIT(3);                       // <=3 outstanding => current tile landed
        } else {
            ASYNC_WAIT(0);
        }

        float4 d4 = sd[w][buf][lane];
        int4   ia = si[w][buf][2 * lane];        // region ints 8L..8L+3
        int4   ib = si[w][buf][2 * lane + 1];    // region ints 8L+4..8L+7
        // pair j of this lane: col0 at region int 8L+2j -> .x/.z components
        body(d4.x, ia.x);
        body(d4.y, ia.z);
        body(d4.z, ib.x);
        body(d4.w, ib.z);

        buf ^= 1;
    }

    // scalar tail for npairs not divisible by TILE_PAIRS (empty for 6.4M)
    const int tail0  = ntiles * TILE_PAIRS;
    const int stride = gridDim.x * blockDim.x;
    for (int p = tail0 + blockIdx.x * blockDim.x + threadIdx.x; p < npairs; p += stride) {
        body(dist[p], ind2[2 * p]);
    }
}

__global__ void lj_zero_kernel(float* __restrict__ out, int n) {
    int i = blockIdx.x * blockDim.x + threadIdx.x;
    if (i < n) out[i] = 0.0f;
}

extern "C" void kernel_launch(void* const* d_in, const int* in_sizes, int n_in,
                              void* d_out, int out_size, void* d_ws, size_t ws_size,
                              hipStream_t stream) {
    const float* dist = (const float*)d_in[0];
    // d_in[1] (ind_1) is only used for its static shape in the reference: dead data.
    const int*   ind2 = (const int*)d_in[2];
    float*       out  = (float*)d_out;

    const int npairs = in_sizes[0];
    const int natoms = out_size;
    const int ntiles = npairs / TILE_PAIRS;

    lj_zero_kernel<<<(natoms + 255) / 256, 256, 0, stream>>>(out, natoms);
    lj_scatter_kernel<<<1024, 256, 0, stream>>>(dist, ind2, out, npairs, ntiles);
}